// Visibility_11433202942091
// MI455X (gfx1250) — compile-verified
//
#include <hip/hip_runtime.h>

typedef __attribute__((ext_vector_type(2))) float v2f;
typedef __attribute__((ext_vector_type(8))) float v8f;

#define CC 16
#define RR 1024
#define TT 4096

// ---------------------------------------------------------------------------
// Kernel 1: init max-table to -1 (65,536 entries in d_ws)
// ---------------------------------------------------------------------------
__global__ __launch_bounds__(256) void vis_init(int* __restrict__ m) {
    int i = blockIdx.x * 256 + threadIdx.x;
    m[i] = -1;
}

// ---------------------------------------------------------------------------
// Kernel 2: scatter max-rho per (f, theta) column.
// 500k atomics into a 256KB L2-resident table -> negligible cost.
// ---------------------------------------------------------------------------
__global__ __launch_bounds__(256) void vis_scatter(const int* __restrict__ rho,
                                                   const int* __restrict__ theta,
                                                   const int* __restrict__ f,
                                                   int* __restrict__ m, int n) {
    int i = blockIdx.x * 256 + threadIdx.x;
    if (i < n) {
        atomicMax(m + f[i] * TT + theta[i], rho[i]);
    }
}

// ---------------------------------------------------------------------------
// Kernel 3: expansion. out[c, rr, t] = rp[mhat(c,t)] - r[rr], 256 MB of stores.
// Each 16x16 tile is D = A x B via V_WMMA_F32_16X16X4_F32 with K=2:
//   A[i][0]=1, A[i][1]=-r[rr0+i];  B[0][j]=rp[mhat(t0+j)], B[1][j]=1
// VGPR layouts (ISA 7.12.2, 32-bit 16x4 A / 4x16 B / 16x16 D):
//   A: lanes 0-15 = M, a.x=K0, a.y=K1 ; lanes 16-31 carry K2/K3 -> 0
//   B: lanes 0-15 = N, b.x=K0 row, b.y=K1 row ; lanes 16-31 = K2/K3 rows -> 0
//   D: VGPR v, lanes 0-15 -> row rr0+v, lanes 16-31 -> row rr0+v+8
// One wave per (c, t-tile, chunk-of-16-row-tiles): B loop-invariant, 16 WMMAs,
// 128 coalesced b32 stores (each = two contiguous 64B row segments).
// ---------------------------------------------------------------------------
__global__ __launch_bounds__(256) void vis_expand(const int* __restrict__ m,
                                                  const float* __restrict__ rp,
                                                  const float* __restrict__ r,
                                                  float* __restrict__ out) {
    const int lane  = threadIdx.x & 31;
    const int wid   = blockIdx.x * 8 + (threadIdx.x >> 5);
    const int ttile = wid & 255;         // 256 t-tiles of 16
    const int chunk = (wid >> 8) & 3;    // 4 chunks x 16 row-tiles = 1024 rows
    const int c     = wid >> 10;         // 16 channels
    const int t0    = ttile << 4;
    const int rrb   = chunk << 8;        // first row of this chunk
    const int n     = lane & 15;
    const bool hi   = lane >= 16;

    // B matrix (loop-invariant): K0 row = column maxima, K1 row = ones.
    float bx = 0.0f, by = 0.0f, ax = 0.0f;
    if (!hi) {
        int v = m[c * TT + t0 + n];
        bx = rp[v < 0 ? RR : v];         // empty column -> rp[R]
        by = 1.0f;
        ax = 1.0f;                        // A K0 column = ones
    }
    v2f b; b.x = bx; b.y = by;
    v2f a; a.x = ax; a.y = 0.0f;

    float* base = out + ((size_t)c * RR + rrb + (hi ? 8 : 0)) * TT + t0 + n;

    #pragma unroll 4
    for (int tile = 0; tile < 16; ++tile) {
        const int rr0 = rrb + (tile << 4);
        const float rv = r[rr0 + n];      // row offsets for M = 0..15
        a.y = hi ? 0.0f : -rv;            // A K1 column = -r[row]

        v8f cz = {};
        v8f d = __builtin_amdgcn_wmma_f32_16x16x4_f32(
            /*neg_a=*/false, a, /*neg_b=*/false, b,
            /*c_mod=*/(short)0, cz, /*reuse_a=*/false, /*reuse_b=*/false);

        float* p = base + (size_t)tile * (16 * TT);
        #pragma unroll
        for (int v = 0; v < 8; ++v) {
            p[v * TT] = d[v];             // two 64B contiguous segments / store
        }
    }
}

// ---------------------------------------------------------------------------
extern "C" void kernel_launch(void* const* d_in, const int* in_sizes, int n_in,
                              void* d_out, int out_size, void* d_ws, size_t ws_size,
                              hipStream_t stream) {
    (void)n_in; (void)out_size; (void)ws_size;
    const int*   rho   = (const int*)d_in[0];
    const int*   theta = (const int*)d_in[1];
    const int*   f     = (const int*)d_in[2];
    const float* rp    = (const float*)d_in[3];
    const float* r     = (const float*)d_in[4];
    float*       out   = (float*)d_out;
    const int    N     = in_sizes[0];

    int* m = (int*)d_ws;  // 16*4096 ints = 256 KB scratch

    vis_init<<<(CC * TT) / 256, 256, 0, stream>>>(m);
    vis_scatter<<<(N + 255) / 256, 256, 0, stream>>>(rho, theta, f, m, N);
    // 16 c * 256 t-tiles * 4 chunks = 16384 waves = 2048 blocks of 8 waves
    vis_expand<<<2048, 256, 0, stream>>>(m, rp, r, out);
}